// ChamferDistanceLoss_20109036880109
// MI455X (gfx1250) — compile-verified
//
#include <hip/hip_runtime.h>

// Problem geometry (compile-time constants from the reference)
#define R_D 256
#define A_D 128
#define E_D 32
#define NB  4
#define S_PER_B (R_D * A_D * E_D)      // 1,048,576 elements per batch
#define NTOT (NB * S_PER_B)            // 4,194,304 total elements

typedef float v2f __attribute__((ext_vector_type(2)));
typedef float v8f __attribute__((ext_vector_type(8)));

// Wave32 sum reduction on the matrix pipe:
// A (16x4 f32) holds the 32 lane partials (lanes 0-15 in K=0, lanes 16-31 in K=2),
// B = ones => D[m][n] = s_m + s_{m+16}. Summing the 8 D VGPRs gives the low-half
// row-sum total in lanes 0-15 and the high-half total in lanes 16-31; one xor-16
// shuffle-add produces the full wave sum in every lane.
__device__ __forceinline__ float wave_sum_wmma(float s) {
  v2f a; a[0] = s;    a[1] = 0.0f;
  v2f b; b[0] = 1.0f; b[1] = 1.0f;
  v8f c = {};
  c = __builtin_amdgcn_wmma_f32_16x16x4_f32(
      /*neg_a=*/false, a, /*neg_b=*/false, b,
      /*c_mod=*/(short)0, c, /*reuse_a=*/false, /*reuse_b=*/false);
  float t = 0.0f;
#pragma unroll
  for (int k = 0; k < 8; ++k) t += c[k];
  t += __shfl_xor(t, 16, 32);
  return t;
}

// Block (256 threads = 8 waves) sum; result valid in thread 0.
__device__ __forceinline__ float block_sum(float s) {
  __shared__ float sm[8];
  float w = wave_sum_wmma(s);
  const int lane = threadIdx.x & 31;
  const int wv   = threadIdx.x >> 5;
  if (lane == 0) sm[wv] = w;
  __syncthreads();
  float r = 0.0f;
  if (threadIdx.x == 0) {
#pragma unroll
    for (int k = 0; k < 8; ++k) r += sm[k];
  }
  return r;
}

// Kernel 1: p2[b,s] = sigmoid(pcl2[b,1,s] - pcl2[b,0,s]), 4 elements/thread (B128 loads/stores).
__global__ void __launch_bounds__(256)
p2_sigmoid_kernel(const float* __restrict__ pcl2, float* __restrict__ p2) {
  const long i = ((long)blockIdx.x * 256 + threadIdx.x) * 4;
  const int b = (int)(i >> 20);
  const int s = (int)(i & (S_PER_B - 1));
  const float* base = pcl2 + (size_t)b * 2 * S_PER_B + s;
  float4 c0 = *(const float4*)(base);
  float4 c1 = *(const float4*)(base + S_PER_B);
  float4 o;
  o.x = 1.0f / (1.0f + __expf(c0.x - c1.x));
  o.y = 1.0f / (1.0f + __expf(c0.y - c1.y));
  o.z = 1.0f / (1.0f + __expf(c0.z - c1.z));
  o.w = 1.0f / (1.0f + __expf(c0.w - c1.w));
  *(float4*)(p2 + (size_t)b * S_PER_B + s) = o;
}

// Kernel 2: trilinear border-clamped gather of p2 at flow-warped coordinates,
// fused with the clamped-log BCE term; WMMA wave/block reduction to partials.
__global__ void __launch_bounds__(256)
warp_bce_kernel(const float* __restrict__ pcl1,
                const float* __restrict__ flow,
                const float* __restrict__ p2,
                float* __restrict__ partials) {
  const long i = ((long)blockIdx.x * 256 + threadIdx.x) * 4;
  const int b  = (int)(i >> 20);
  const int s  = (int)(i & (S_PER_B - 1));
  const int e0 = s & (E_D - 1);          // 4 consecutive e's share r,a
  const int a  = (s >> 5) & (A_D - 1);
  const int r  = s >> 12;

  const float* pb = pcl1 + (size_t)b * 2 * S_PER_B + s;
  float4 q0 = *(const float4*)(pb);
  float4 q1 = *(const float4*)(pb + S_PER_B);
  const float* fb = flow + (size_t)b * 3 * S_PER_B + s;
  float4 f0 = *(const float4*)(fb);                 // flow channel 0 -> x (E axis), *31/255 scale
  float4 f1 = *(const float4*)(fb + S_PER_B);       // channel 1 -> y (A axis), unit scale
  float4 f2 = *(const float4*)(fb + 2 * S_PER_B);   // channel 2 -> z (R axis), *255/31 scale
  const float* p2b = p2 + (size_t)b * S_PER_B;

  const float* q0p = (const float*)&q0;
  const float* q1p = (const float*)&q1;
  const float* f0p = (const float*)&f0;
  const float* f1p = (const float*)&f1;
  const float* f2p = (const float*)&f2;

  float acc = 0.0f;
#pragma unroll
  for (int j = 0; j < 4; ++j) {
    // Exact algebraic reduction of the reference's normalize->unnormalize chain
    float xx = fminf(fmaxf((float)(e0 + j) + f0p[j] * (31.0f / 255.0f), 0.0f), (float)(E_D - 1));
    float yy = fminf(fmaxf((float)a        + f1p[j],                    0.0f), (float)(A_D - 1));
    float zz = fminf(fmaxf((float)r        + f2p[j] * (255.0f / 31.0f), 0.0f), (float)(R_D - 1));
    int x0 = (int)xx, y0 = (int)yy, z0 = (int)zz;
    float wx = xx - (float)x0, wy = yy - (float)y0, wz = zz - (float)z0;
    int x1  = min(x0 + 1, E_D - 1);
    int y0o = y0 << 5, y1o = min(y0 + 1, A_D - 1) << 5;
    int z0o = z0 << 12, z1o = min(z0 + 1, R_D - 1) << 12;

    float g000 = p2b[z0o + y0o + x0], g001 = p2b[z0o + y0o + x1];
    float g010 = p2b[z0o + y1o + x0], g011 = p2b[z0o + y1o + x1];
    float g100 = p2b[z1o + y0o + x0], g101 = p2b[z1o + y0o + x1];
    float g110 = p2b[z1o + y1o + x0], g111 = p2b[z1o + y1o + x1];

    float c00 = g000 + wx * (g001 - g000);
    float c01 = g010 + wx * (g011 - g010);
    float c10 = g100 + wx * (g101 - g100);
    float c11 = g110 + wx * (g111 - g110);
    float cl0 = c00 + wy * (c01 - c00);
    float cl1 = c10 + wy * (c11 - c10);
    float w   = cl0 + wz * (cl1 - cl0);   // warped_p1

    // log(sigmoid(d)) = min(d,0) - log1p(exp(-|d|)); log(1-sigmoid(d)) = -max(d,0) - same
    float d  = q1p[j] - q0p[j];
    float sp = __logf(1.0f + __expf(-fabsf(d)));
    float logp   = fmaxf(fminf(d, 0.0f) - sp,  -100.0f);
    float log1mp = fmaxf(-fmaxf(d, 0.0f) - sp, -100.0f);
    acc += w * logp + (1.0f - w) * log1mp;
  }

  float bsum = block_sum(acc);
  if (threadIdx.x == 0) partials[blockIdx.x] = bsum;
}

// Kernel 3: deterministic reduction of the 4096 block partials -> -mean.
__global__ void __launch_bounds__(256)
finalize_kernel(const float* __restrict__ partials, float* __restrict__ out) {
  float s = 0.0f;
#pragma unroll
  for (int k = 0; k < 16; ++k) s += partials[k * 256 + threadIdx.x];
  float tot = block_sum(s);
  if (threadIdx.x == 0) out[0] = -tot * (1.0f / (float)NTOT);
}

extern "C" void kernel_launch(void* const* d_in, const int* in_sizes, int n_in,
                              void* d_out, int out_size, void* d_ws, size_t ws_size,
                              hipStream_t stream) {
  (void)in_sizes; (void)n_in; (void)out_size; (void)ws_size;
  const float* pcl1 = (const float*)d_in[0];
  const float* pcl2 = (const float*)d_in[1];
  const float* flow = (const float*)d_in[2];

  float* p2       = (float*)d_ws;                                      // 16 MB (fits in L2)
  float* partials = (float*)((char*)d_ws + (size_t)NTOT * sizeof(float)); // +16 KB

  const int threads = 256;                       // 8 wave32 per block
  const int blocks  = NTOT / 4 / threads;        // 4096 blocks, exact cover, no tail
  p2_sigmoid_kernel<<<blocks, threads, 0, stream>>>(pcl2, p2);
  warp_bce_kernel<<<blocks, threads, 0, stream>>>(pcl1, flow, p2, partials);
  finalize_kernel<<<1, threads, 0, stream>>>(partials, (float*)d_out);
}